// CorollaryResonanceBank_7284264534656
// MI455X (gfx1250) — compile-verified
//
#include <hip/hip_runtime.h>
#include <math.h>

typedef float v2f __attribute__((ext_vector_type(2)));
typedef float v8f __attribute__((ext_vector_type(8)));

#define B_   256
#define C_   64
#define T_   2048
#define R_   16
#define TB_  128   // TIME_BINS
// window = T_/TB_ = 16

__global__ __launch_bounds__(256)
void snn_resonance_fused(const float* __restrict__ tx,     // (B,C,T)
                         const float* __restrict__ rx,     // (B,2,C,T)
                         const float* __restrict__ conv_w, // (16,3)
                         const float* __restrict__ conv_b, // (16,)
                         const float* __restrict__ freq_p, // (16,)
                         const float* __restrict__ dec_p,  // (16,)
                         const float* __restrict__ thr_p,  // (16,)
                         const float* __restrict__ tw_p,   // scalar
                         const float* __restrict__ rw_p,   // scalar
                         float* __restrict__ out)          // (B,16)
{
    __shared__ float traces[3][TB_];   // signed_distance, azimuth, elevation
    __shared__ float drive[R_][TB_];   // post-conv drive, [r][bin]
    __shared__ float red[8];
    __shared__ float s_scale;

    const int b    = blockIdx.x;
    const int tid  = threadIdx.x;
    const int wave = tid >> 5;
    const int lane = tid & 31;

    // softplus of scalar weights
    const float sp_tw = log1pf(expf(tw_p[0]));
    const float sp_rw = log1pf(expf(rw_p[0]));

    const float* txb = tx  + (size_t)b * C_ * T_;
    const float* rx0 = rx  + (size_t)b * 2 * C_ * T_;
    const float* rx1 = rx0 + (size_t)C_ * T_;

    // ---- Stage 1: channel reduction + 16-wide time pooling (bandwidth bound) ----
    // Each wave handles an 8-bin block (128 consecutive t): lane covers t = blk*128 + lane*4 .. +3,
    // which lies entirely inside bin (blk*8 + lane/4). 512B contiguous per channel row per wave.
    for (int blk = wave; blk < 16; blk += 8) {
        const int t0 = blk * 128 + lane * 4;
        float accT = 0.f, accL = 0.f, accR = 0.f, accE = 0.f;
        #pragma unroll 4
        for (int c = 0; c < C_; ++c) {
            const float axis = -1.0f + (2.0f / 63.0f) * (float)c;  // linspace(-1,1,64)
            float4 a = *(const float4*)(txb + (size_t)c * T_ + t0);
            float4 l = *(const float4*)(rx0 + (size_t)c * T_ + t0);
            float4 r = *(const float4*)(rx1 + (size_t)c * T_ + t0);
            float sa = (a.x + a.y) + (a.z + a.w);
            float sl = (l.x + l.y) + (l.z + l.w);
            float sr = (r.x + r.y) + (r.z + r.w);
            accT += sa;
            accL += sl;
            accR += sr;
            accE += axis * (sl + sr);   // elevation sums both ears
        }
        // reduce the 4 lanes that share a bin
        #pragma unroll
        for (int m = 1; m < 4; m <<= 1) {
            accT += __shfl_xor(accT, m, 32);
            accL += __shfl_xor(accL, m, 32);
            accR += __shfl_xor(accR, m, 32);
            accE += __shfl_xor(accE, m, 32);
        }
        if ((lane & 3) == 0) {
            const int bin = blk * 8 + (lane >> 2);
            const float inv_win = 1.0f / 16.0f;     // mean over window
            traces[0][bin] = (sp_rw * 0.5f * (accL + accR) - sp_tw * accT) * inv_win;
            traces[1][bin] = (accL - accR) * inv_win;
            traces[2][bin] = accE * inv_win;
        }
    }
    __syncthreads();

    // ---- Stage 2: per-batch scale = max(max|traces|, 1) ----
    {
        float m = 0.0f;
        const float* tp = &traces[0][0];
        for (int i = tid; i < 3 * TB_; i += 256) m = fmaxf(m, fabsf(tp[i]));
        #pragma unroll
        for (int s = 16; s > 0; s >>= 1) m = fmaxf(m, __shfl_xor(m, s, 32));
        if (lane == 0) red[wave] = m;
        __syncthreads();
        if (tid == 0) {
            float mm = red[0];
            #pragma unroll
            for (int i = 1; i < 8; ++i) mm = fmaxf(mm, red[i]);
            s_scale = fmaxf(mm, 1.0f);
        }
        __syncthreads();
    }
    const float inv_scale = 1.0f / s_scale;

    // ---- Stage 3: 1x1 conv via V_WMMA_F32_16X16X4_F32 ----
    // D(16x16) = A(16x4: conv_w, K=3 zero-padded) x B(4x16: traces tile) + C(conv_b)
    // f32 fragment layout (wave32): lanes 0-15 = K-half 0 / rows M=0..7, lanes 16-31 = K-half 1 / rows M=8..15.
    {
        const int half = lane >> 4;       // 0: K={0,1}, 1: K={2,3}
        const int l16  = lane & 15;

        // A fragment: row M=l16, cols K=2*half+{0,1}; K=3 padded with 0
        const float w0 = conv_w[l16 * 3 + 0];
        const float w1 = conv_w[l16 * 3 + 1];
        const float w2 = conv_w[l16 * 3 + 2];
        v2f a;
        a.x = half ? w2 : w0;
        a.y = half ? 0.0f : w1;

        // B fragment: col N=l16 of this wave's 16-bin tile, rows K=2*half+{0,1}
        const int bin = wave * 16 + l16;
        const float b0v = traces[0][bin];
        const float b1v = traces[1][bin];
        const float b2v = traces[2][bin];
        v2f bb;
        bb.x = (half ? b2v : b0v) * inv_scale;
        bb.y = half ? 0.0f : (b1v * inv_scale);

        // C fragment: conv_b broadcast along N; VGPR i holds row M = i + 8*half
        v8f cc;
        #pragma unroll
        for (int i = 0; i < 8; ++i) cc[i] = conv_b[i + half * 8];

        v8f d = __builtin_amdgcn_wmma_f32_16x16x4_f32(
            /*neg_a=*/false, a, /*neg_b=*/false, bb,
            /*c_mod=*/(short)0, cc, /*reuse_a=*/false, /*reuse_b=*/false);

        // scatter D (time across lanes) into LDS as drive[r][bin] for the sequential scan
        #pragma unroll
        for (int i = 0; i < 8; ++i) drive[i + half * 8][bin] = d[i];
    }
    __syncthreads();

    // ---- Stage 4: resonate-and-fire scan over 128 bins (lane = resonator r) ----
    if (tid < R_) {
        const int r = tid;
        const float freq = 0.02f + 0.18f / (1.0f + expf(-freq_p[r]));
        const float dec  = 0.80f + 0.18f / (1.0f + expf(-dec_p[r]));
        const float thr  = 0.35f + 0.75f / (1.0f + expf(-thr_p[r]));
        float state = 0.0f, vel = 0.0f, acc = 0.0f;
        #pragma unroll 8
        for (int t = 0; t < TB_; ++t) {
            const float cur = drive[r][t];
            vel   = dec * vel + cur - freq * state;
            state = state + freq * vel;
            const float spike = (state - thr) > 0.0f ? 1.0f : 0.0f;
            state -= spike * thr;
            acc   += spike;
        }
        out[b * R_ + r] = acc * (1.0f / (float)TB_);
    }
}

extern "C" void kernel_launch(void* const* d_in, const int* in_sizes, int n_in,
                              void* d_out, int out_size, void* d_ws, size_t ws_size,
                              hipStream_t stream) {
    (void)in_sizes; (void)n_in; (void)out_size; (void)d_ws; (void)ws_size;
    const float* tx     = (const float*)d_in[0];  // transmit_spikes (B,C,T)
    const float* rx     = (const float*)d_in[1];  // receive_spikes  (B,2,C,T)
    const float* conv_w = (const float*)d_in[2];  // (16,3)
    const float* conv_b = (const float*)d_in[3];  // (16,)
    const float* freq   = (const float*)d_in[4];  // (16,)
    const float* dec    = (const float*)d_in[5];  // (16,)
    const float* thr    = (const float*)d_in[6];  // (16,)
    const float* tw     = (const float*)d_in[7];  // scalar
    const float* rw     = (const float*)d_in[8];  // scalar
    float* out = (float*)d_out;                   // (B,16)

    snn_resonance_fused<<<B_, 256, 0, stream>>>(tx, rx, conv_w, conv_b,
                                                freq, dec, thr, tw, rw, out);
}